// HybridQLSTM_65481071405950
// MI455X (gfx1250) — compile-verified
//
#include <hip/hip_runtime.h>
#include <hip/hip_bf16.h>

// ---------------------------------------------------------------------------
// HybridQLSTM for MI455X (gfx1250, wave32).
//
//   conv[t,b]    = mean_i( prefixXOR_i( x[t,b,:] > 0 ) )
//   raw[t,b,g,q] = x[t,b,:].Wg[q,0:256] + conv[t,b]*sum(Wg[q,256:512]) + bg[q]
//   gate[t,b,g]  = mean_q( prefixXOR_q( raw > 0.5 ) )
//   c_t[b,h]     = F_t[b]*cx0[b,h] + s_t[b]   (F=prod f; gates are row scalars)
//   outs[t,b,h]  = o_t[b]*tanh(c_t[b,h])
//
// Dominant work: one f32 GEMM (T*B=131072) x (4Q=1024), K=256 on the matrix
// pipe (V_WMMA_F32_16X16X4_F32; f32 required by the hard >0.5 threshold).
// v2: 2 row-tiles/WG (halves W L2 traffic, 2 independent WMMA chains),
// conflict-free b64 A-fragment loads from a 260-stride row-major LDS tile,
// and ballot-packed threshold bits (4 KB LDS instead of 66 KB raw floats).
// ---------------------------------------------------------------------------

#define T_STEPS 512
#define BATCH   256
#define DIM     256
#define HID     256
#define QB      256
#define TB      (T_STEPS * BATCH)   // 131072 rows
#define NCOL    (4 * QB)            // 1024 gate columns
#define XS_LD   260                 // padded A-tile row stride (floats)

typedef __attribute__((ext_vector_type(2))) float v2f;
typedef __attribute__((ext_vector_type(8))) float v8f;

// ---- wave32 prefix-XOR parity fraction over 256 bits (8 bits per lane) ----
__device__ __forceinline__ float parity_frac_wave(const float* v, float thresh) {
  unsigned bits = 0u;
#pragma unroll
  for (int j = 0; j < 8; ++j) bits |= (v[j] > thresh ? 1u : 0u) << j;
  unsigned p = bits;                 // p bit j = XOR of bits 0..j
  p ^= p << 1; p ^= p << 2; p ^= p << 4;
  int localsum = __popc(p & 0xFFu);
  unsigned laneXor = (p >> 7) & 1u;
  unsigned lane = threadIdx.x & 31u;
  unsigned long long bal = __ballot(laneXor != 0u);
  unsigned carry = (unsigned)__popcll(bal & ((1ull << lane) - 1ull)) & 1u;
  int s = carry ? (8 - localsum) : localsum;
#pragma unroll
  for (int off = 16; off; off >>= 1) s += __shfl_xor(s, off, 32);
  return (float)s * (1.0f / 256.0f);
}

// ---- phase 0: wsum[g*256+q] = sum_d W_g[q, 256+d] (wave per output) -------
__global__ __launch_bounds__(256) void qlstm_wsum_kernel(
    const float* __restrict__ Wf, const float* __restrict__ Wi,
    const float* __restrict__ Wu, const float* __restrict__ Wo,
    float* __restrict__ wsum) {
  const int lane = threadIdx.x & 31;
  const int idx  = blockIdx.x * 8 + (threadIdx.x >> 5);   // 0..1023
  const int g = idx >> 8, q = idx & 255;
  const float* W = (g == 0) ? Wf : (g == 1) ? Wi : (g == 2) ? Wu : Wo;
  const float* p = W + (size_t)q * 512 + 256 + lane * 8;
  float4 a = *(const float4*)p;
  float4 b = *(const float4*)(p + 4);
  float s = a.x + a.y + a.z + a.w + b.x + b.y + b.z + b.w;
#pragma unroll
  for (int off = 16; off; off >>= 1) s += __shfl_xor(s, off, 32);
  if (lane == 0) wsum[idx] = s;
}

// ---- phase 1: conv[row] (wave per row) ------------------------------------
__global__ __launch_bounds__(256) void qlstm_conv_kernel(
    const float* __restrict__ x, float* __restrict__ conv) {
  const int lane = threadIdx.x & 31;
  const int row  = blockIdx.x * 8 + (threadIdx.x >> 5);
  const float* p = x + (size_t)row * DIM + lane * 8;
  float v[8];
  float4 a = *(const float4*)p;
  float4 b = *(const float4*)(p + 4);
  v[0]=a.x; v[1]=a.y; v[2]=a.z; v[3]=a.w; v[4]=b.x; v[5]=b.y; v[6]=b.z; v[7]=b.w;
  float frac = parity_frac_wave(v, 0.0f);
  if (lane == 0) conv[row] = frac;
}

// ---- phase 2: fused GEMM (f32 WMMA, 2 row-tiles) + threshold + parity -----
// Workgroup: 32 rows x 1024 cols, K=256. 8 waves x 8 column-tiles, each
// column-tile runs two WMMA accumulation chains (one per row-tile) sharing
// the same B fragments.
__global__ __launch_bounds__(256) void qlstm_gates_kernel(
    const float* __restrict__ x,
    const float* __restrict__ Wf, const float* __restrict__ bf,
    const float* __restrict__ Wi, const float* __restrict__ bi,
    const float* __restrict__ Wu, const float* __restrict__ bu,
    const float* __restrict__ Wo, const float* __restrict__ bo,
    const float* __restrict__ conv, const float* __restrict__ wsum,
    float* __restrict__ gates) {
  __shared__ float xs[2][16 * XS_LD];      // row-major A tiles, pad 260
  __shared__ unsigned bitsW[64 * 2 * 8];   // [ct][tt][r] ballot masks (4 KB)
  __shared__ float conv_s[32];

  const int tid  = threadIdx.x;
  const int lane = tid & 31;
  const int wave = tid >> 5;
  const int r0   = blockIdx.x * 32;        // first row (t*B+b) of this block

  // Stage both A tiles: contiguous 16-float (b128 x4) copies per thread.
  {
    const int m  = tid >> 4;               // row within tile 0..15
    const int kb = (tid & 15) << 4;        // k base 0..240
#pragma unroll
    for (int tt = 0; tt < 2; ++tt) {
      const float4* src =
          (const float4*)(x + (size_t)(r0 + tt * 16 + m) * DIM + kb);
      float4* dst = (float4*)&xs[tt][m * XS_LD + kb];
#pragma unroll
      for (int vq = 0; vq < 4; ++vq) dst[vq] = src[vq];
    }
  }
  if (tid < 32) conv_s[tid] = conv[r0 + tid];
  __syncthreads();

  const int hi = lane >> 4;                // 0: lanes 0-15 (K+0,1), 1: (K+2,3)
  const int ln = lane & 15;
  const int abase = ln * XS_LD + 2 * hi;   // A frag: row=ln, k=4kk+2hi (+0,+1)

#pragma unroll 1
  for (int i = 0; i < 8; ++i) {
    const int ct  = wave * 8 + i;          // column tile 0..63 (gate uniform)
    const int g   = ct >> 4;
    const int col = ct * 16 + ln;          // 0..1023
    const int q   = col & 255;
    const float* Wg = (g == 0) ? Wf : (g == 1) ? Wi : (g == 2) ? Wu : Wo;
    const float* Bg = (g == 0) ? bf : (g == 1) ? bi : (g == 2) ? bu : bo;
    const float* wrow = Wg + (size_t)q * 512 + 2 * hi;

    v8f acc0 = {}, acc1 = {};
#pragma unroll 8
    for (int kk = 0; kk < 64; ++kk) {      // K = 64 * 4
      const float2 bw = *(const float2*)(wrow + kk * 4);
      v2f b; b.x = bw.x; b.y = bw.y;
      const v2f a0 = *(const v2f*)&xs[0][abase + kk * 4];
      const v2f a1 = *(const v2f*)&xs[1][abase + kk * 4];
      acc0 = __builtin_amdgcn_wmma_f32_16x16x4_f32(
          false, a0, false, b, (short)0, acc0, false, false);
      acc1 = __builtin_amdgcn_wmma_f32_16x16x4_f32(
          false, a1, false, b, (short)0, acc1, false, false);
    }

    const float cw   = wsum[col];
    const float bias = Bg[q];
#pragma unroll
    for (int r = 0; r < 8; ++r) {          // C layout: VGPR r -> M = r + 8*hi
      const unsigned m0 = (unsigned)__ballot(
          acc0[r] + conv_s[r + 8 * hi] * cw + bias > 0.5f);
      const unsigned m1 = (unsigned)__ballot(
          acc1[r] + conv_s[16 + r + 8 * hi] * cw + bias > 0.5f);
      if (lane == 0) {                     // bits 0-15: row r; 16-31: row r+8
        bitsW[(ct * 2 + 0) * 8 + r] = m0;
        bitsW[(ct * 2 + 1) * 8 + r] = m1;
      }
    }
  }
  __syncthreads();

  // 128 (row,gate) parity tasks, one thread each; 16-bit prefix-XOR chain.
  if (tid < 128) {
    const int rowL = tid >> 2;             // 0..31
    const int g    = tid & 3;
    const int tt   = rowL >> 4;
    const int m    = rowL & 15;
    const unsigned short* bits16 = (const unsigned short*)bitsW;
    const int sub = ((m & 7) << 1) + (m >> 3);   // u16 slot within block
    int total = 0;
    unsigned carry = 0;
#pragma unroll
    for (int cg = 0; cg < 16; ++cg) {      // q ascending: ct = g*16 + cg
      const int ct = g * 16 + cg;
      unsigned p = bits16[(ct * 2 + tt) * 16 + sub];
      p ^= p << 1; p ^= p << 2; p ^= p << 4; p ^= p << 8;
      p &= 0xFFFFu;
      const int ls = __popc(p);
      total += carry ? (16 - ls) : ls;
      carry ^= (p >> 15) & 1u;
    }
    gates[(size_t)(r0 + tt * 16 + m) * 4 + g] = (float)total * (1.0f / 256.0f);
  }
}

// ---- phase 3: scalar scan per batch row -----------------------------------
__global__ __launch_bounds__(256) void qlstm_scan_kernel(
    const float* __restrict__ gates, float* __restrict__ s_arr,
    float* __restrict__ F_arr, float* __restrict__ o_arr) {
  const int b = threadIdx.x;
  float s = 0.0f, F = 1.0f;
  for (int t = 0; t < T_STEPS; ++t) {
    const float4 gv = *(const float4*)(gates + ((size_t)t * BATCH + b) * 4);
    const float f  = 1.0f / (1.0f + expf(-gv.x));
    const float ii = 1.0f / (1.0f + expf(-gv.y));
    const float g  = tanhf(gv.z);
    const float o  = 1.0f / (1.0f + expf(-gv.w));
    s = f * s + ii * g;
    F = F * f;
    const size_t idx = (size_t)t * BATCH + b;
    s_arr[idx] = s;  F_arr[idx] = F;  o_arr[idx] = o;
  }
}

// ---- phase 4: broadcast to outs (T,B,H) ++ hx (B,H) ++ cx (B,H) -----------
__global__ __launch_bounds__(256) void qlstm_bcast_kernel(
    const float* __restrict__ s_arr, const float* __restrict__ F_arr,
    const float* __restrict__ o_arr, const float* __restrict__ cx0,
    float4* __restrict__ out) {
  const size_t OUT4 = (size_t)TB * (HID / 4);
  const size_t HX4  = (size_t)BATCH * (HID / 4);
  const size_t idx  = (size_t)blockIdx.x * 256 + threadIdx.x;
  if (idx >= OUT4 + 2 * HX4) return;

  size_t row;
  bool is_cx = false;
  if (idx < OUT4) {
    row = idx >> 6;                        // 64 float4 per row
  } else if (idx < OUT4 + HX4) {
    row = (size_t)(T_STEPS - 1) * BATCH + ((idx - OUT4) >> 6);
  } else {
    row = (size_t)(T_STEPS - 1) * BATCH + ((idx - OUT4 - HX4) >> 6);
    is_cx = true;
  }
  const int b  = (int)(row & (BATCH - 1));
  const int h4 = ((int)(idx & 63)) * 4;    // regions are 64-float4 aligned
  const float4 c0 = *(const float4*)(cx0 + (size_t)b * HID + h4);
  const float sv = s_arr[row], Fv = F_arr[row], ov = o_arr[row];
  float4 r;
  if (!is_cx) {
    r.x = ov * tanhf(Fv * c0.x + sv);
    r.y = ov * tanhf(Fv * c0.y + sv);
    r.z = ov * tanhf(Fv * c0.z + sv);
    r.w = ov * tanhf(Fv * c0.w + sv);
  } else {
    r.x = Fv * c0.x + sv;  r.y = Fv * c0.y + sv;
    r.z = Fv * c0.z + sv;  r.w = Fv * c0.w + sv;
  }
  out[idx] = r;
}

extern "C" void kernel_launch(void* const* d_in, const int* in_sizes, int n_in,
                              void* d_out, int out_size, void* d_ws, size_t ws_size,
                              hipStream_t stream) {
  const float* x   = (const float*)d_in[0];
  // d_in[1] = hx0 (outputs never depend on it)
  const float* cx0 = (const float*)d_in[2];
  const float* Wf = (const float*)d_in[3];  const float* bf = (const float*)d_in[4];
  const float* Wi = (const float*)d_in[5];  const float* bi = (const float*)d_in[6];
  const float* Wu = (const float*)d_in[7];  const float* bu = (const float*)d_in[8];
  const float* Wo = (const float*)d_in[9];  const float* bo = (const float*)d_in[10];

  float* ws    = (float*)d_ws;
  float* conv  = ws;                       // TB
  float* wsum  = conv + TB;                // 1024
  float* gates = wsum + NCOL;              // TB*4
  float* s_arr = gates + (size_t)TB * 4;   // TB
  float* F_arr = s_arr + TB;               // TB
  float* o_arr = F_arr + TB;               // TB   (total ~4.1 MB)

  qlstm_wsum_kernel<<<NCOL / 8, 256, 0, stream>>>(Wf, Wi, Wu, Wo, wsum);
  qlstm_conv_kernel<<<TB / 8, 256, 0, stream>>>(x, conv);
  qlstm_gates_kernel<<<TB / 32, 256, 0, stream>>>(
      x, Wf, bf, Wi, bi, Wu, bu, Wo, bo, conv, wsum, gates);
  qlstm_scan_kernel<<<1, 256, 0, stream>>>(gates, s_arr, F_arr, o_arr);
  const size_t total4 = (size_t)TB * (HID / 4) + 2 * (size_t)BATCH * (HID / 4);
  qlstm_bcast_kernel<<<(unsigned)((total4 + 255) / 256), 256, 0, stream>>>(
      s_arr, F_arr, o_arr, cx0, (float4*)d_out);
}